// GNN2D_72808285601931
// MI455X (gfx1250) — compile-verified
//
#include <hip/hip_runtime.h>
#include <math.h>

typedef float v2f __attribute__((ext_vector_type(2)));
typedef float v4f __attribute__((ext_vector_type(4)));
typedef float v8f __attribute__((ext_vector_type(8)));
typedef int   v4i __attribute__((vector_size(16)));   // matches async-LDS builtin pointee

#define KDIM 512
#define KK   (512*512)
#define CH   64
#define BSZ  4

// Padded LDS row strides (floats): 16B-aligned for b128 writes, bank-conflict-free reads
#define WS_STRIDE 68   // o*68 mod 64 = o*4 -> 16 consecutive rows hit distinct banks
#define XS_STRIDE 40   // rows ca vs ca+2 land on disjoint bank halves

#if defined(__gfx1250__) && __has_builtin(__builtin_amdgcn_global_load_async_to_lds_b128) && __has_builtin(__builtin_amdgcn_s_wait_asynccnt)
#define ASYNC_LDS 1
typedef __attribute__((address_space(1))) v4i* gv4i_p;   // global
typedef __attribute__((address_space(3))) v4i* lv4i_p;   // LDS
#else
#define ASYNC_LDS 0
#endif

// ---------------------------------------------------------------------------
// Reductions over input A: rsA[b,k], csA[b,k], dgA[b,k]
// ---------------------------------------------------------------------------
__global__ __launch_bounds__(512) void reduceA_k(const float* __restrict__ A,
                                                 float* __restrict__ rsA,
                                                 float* __restrict__ csA,
                                                 float* __restrict__ dgA) {
  int b = blockIdx.x;
  int k = threadIdx.x;
  const float* S = A + (size_t)b * KK;
  float r = 0.f, c = 0.f;
  for (int i = 0; i < KDIM; ++i) {
    r += S[(size_t)k * KDIM + i];
    c += S[(size_t)i * KDIM + k];
  }
  rsA[b * KDIM + k] = r;
  csA[b * KDIM + k] = c;
  dgA[b * KDIM + k] = S[(size_t)k * KDIM + k];
}

// ---------------------------------------------------------------------------
// Layer 0 rank-structure precompute: dp0/u0/v0 [b,o,k]  (in-channels == 1)
// ---------------------------------------------------------------------------
__global__ __launch_bounds__(256) void prep0_k(
    const float* __restrict__ P1, const float* __restrict__ P2,
    const float* __restrict__ P3, const float* __restrict__ P5,
    const float* __restrict__ P6, const float* __restrict__ P7,
    const float* __restrict__ P8,
    const float* __restrict__ rsA, const float* __restrict__ csA,
    const float* __restrict__ dgA,
    float* __restrict__ dp, float* __restrict__ u, float* __restrict__ v) {
  int idx = blockIdx.x * 256 + threadIdx.x;   // over BSZ*CH*KDIM
  int k = idx % KDIM;
  int o = (idx / KDIM) % CH;
  int b = idx / (KDIM * CH);
  float d = dgA[b * KDIM + k];
  float r = rsA[b * KDIM + k];
  float c = csA[b * KDIM + k];
  size_t out = ((size_t)b * CH + o) * KDIM + k;
  dp[out] = P1[o] * d + 0.1f * (P2[o] * r + P3[o] * c);
  u[out]  = 0.1f * P6[o] * d + 0.01f * P7[o] * r;
  v[out]  = 0.1f * P5[o] * d + 0.01f * P8[o] * c;
}

// ---------------------------------------------------------------------------
// Layer 0 main: A1[b,o,k,l] = relu( k==l ? dp0 : P4[o]*A + u0[k] + v0[l] )
// ---------------------------------------------------------------------------
__global__ __launch_bounds__(256) void layer0_k(
    const float* __restrict__ A, const float* __restrict__ P4,
    const float* __restrict__ dp, const float* __restrict__ u,
    const float* __restrict__ v, float* __restrict__ A1) {
  int b = blockIdx.y;
  int k = blockIdx.x >> 1;
  int l = (blockIdx.x & 1) * 256 + threadIdx.x;
  float a = A[(size_t)b * KK + (size_t)k * KDIM + l];
  size_t bo = (size_t)b * CH;
#pragma unroll 4
  for (int o = 0; o < CH; ++o) {
    float val;
    if (k == l) val = dp[(bo + o) * KDIM + k];
    else        val = P4[o] * a + u[(bo + o) * KDIM + k] + v[(bo + o) * KDIM + l];
    val = val > 0.f ? val : 0.f;
    A1[(bo + o) * KK + (size_t)k * KDIM + l] = val;
  }
}

// ---------------------------------------------------------------------------
// Per-(b,o) slice reduction: rs[b,o,k], cs[b,o,l], diag[b,o,k], sum, sumsq
// ---------------------------------------------------------------------------
__global__ __launch_bounds__(256) void reduce_bc_k(
    const float* __restrict__ X,
    float* __restrict__ rs, float* __restrict__ cs, float* __restrict__ dg,
    float* __restrict__ sum, float* __restrict__ sumsq) {
  int b = blockIdx.y, o = blockIdx.x;
  const float* S = X + ((size_t)b * CH + o) * KK;
  int t = threadIdx.x;
  __shared__ float red[256];
  float psum = 0.f, psq = 0.f;
  size_t obase = ((size_t)b * CH + o) * KDIM;
  for (int rr = 0; rr < 2; ++rr) {
    int kk = t + rr * 256;
    const float* row = S + (size_t)kk * KDIM;
    float acc = 0.f;
    for (int l = 0; l < KDIM; ++l) { float x = row[l]; acc += x; psq += x * x; }
    psum += acc;
    rs[obase + kk] = acc;
    dg[obase + kk] = row[kk];
  }
  for (int rr = 0; rr < 2; ++rr) {
    int ll = t + rr * 256;
    float acc = 0.f;
    for (int kk = 0; kk < KDIM; ++kk) acc += S[(size_t)kk * KDIM + ll];
    cs[obase + ll] = acc;
  }
  red[t] = psum; __syncthreads();
  for (int st = 128; st > 0; st >>= 1) { if (t < st) red[t] += red[t + st]; __syncthreads(); }
  if (t == 0) sum[b * CH + o] = red[0];
  __syncthreads();
  red[t] = psq; __syncthreads();
  for (int st = 128; st > 0; st >>= 1) { if (t < st) red[t] += red[t + st]; __syncthreads(); }
  if (t == 0) sumsq[b * CH + o] = red[0];
}

// ---------------------------------------------------------------------------
// BN fold: s[c] = w/sqrt(var+eps); t[c] = b - mean*s
// ---------------------------------------------------------------------------
__global__ void bn_fin_k(const float* __restrict__ sum, const float* __restrict__ sq,
                         const float* __restrict__ bw, const float* __restrict__ bb,
                         float* __restrict__ s, float* __restrict__ t) {
  int o = threadIdx.x;
  if (o >= CH) return;
  float m = 0.f, q = 0.f;
  for (int b = 0; b < BSZ; ++b) { m += sum[b * CH + o]; q += sq[b * CH + o]; }
  const float inv = 1.0f / ((float)BSZ * (float)KK);
  float mean = m * inv;
  float var = q * inv - mean * mean;
  float sc = bw[o] * rsqrtf(var + 1e-5f);
  s[o] = sc;
  t[o] = bb[o] - mean * sc;
}

// ---------------------------------------------------------------------------
// Layer 1 rank-structure precompute (BN0 folded in; GEMM bias folded into u)
// ---------------------------------------------------------------------------
__global__ __launch_bounds__(256) void prep1_k(
    const float* __restrict__ P1, const float* __restrict__ P2,
    const float* __restrict__ P3, const float* __restrict__ P4,
    const float* __restrict__ P5, const float* __restrict__ P6,
    const float* __restrict__ P7, const float* __restrict__ P8,
    const float* __restrict__ rs, const float* __restrict__ cs,
    const float* __restrict__ dg,
    const float* __restrict__ s0, const float* __restrict__ t0,
    float* __restrict__ dp, float* __restrict__ u, float* __restrict__ v) {
  int idx = blockIdx.x * 256 + threadIdx.x;
  int k = idx % KDIM;
  int o = (idx / KDIM) % CH;
  int b = idx / (KDIM * CH);
  float accd = 0.f, accu = 0.f, accv = 0.f;
  for (int c = 0; c < CH; ++c) {
    float sc = s0[c], tc = t0[c];
    size_t base = ((size_t)b * CH + c) * KDIM + k;
    float dn = sc * dg[base] + tc;
    float rn = sc * rs[base] + (float)KDIM * tc;
    float cn = sc * cs[base] + (float)KDIM * tc;
    accd += P1[o * CH + c] * dn + 0.1f * (P2[o * CH + c] * rn + P3[o * CH + c] * cn);
    accu += 0.1f * P6[o * CH + c] * dn + 0.01f * P7[o * CH + c] * rn + P4[o * CH + c] * tc;
    accv += 0.1f * P5[o * CH + c] * dn + 0.01f * P8[o * CH + c] * cn;
  }
  size_t out = ((size_t)b * CH + o) * KDIM + k;
  dp[out] = accd; u[out] = accu; v[out] = accv;
}

__global__ void foldW_k(const float* __restrict__ P4, const float* __restrict__ s0,
                        float* __restrict__ Wf) {
  int i = blockIdx.x * 256 + threadIdx.x;
  if (i < CH * CH) Wf[i] = P4[i] * s0[i % CH];
}

// ---------------------------------------------------------------------------
// Layer 1 flagship: FP32 WMMA 16x16x4 with async global->LDS operand staging.
// Workgroup: row k, 32 columns. Stage W (64x64) + X (64x32) into padded LDS
// via GLOBAL_LOAD_ASYNC_TO_LDS_B128 (ASYNCcnt), then 8 waves (4 o-tiles x
// 2 n-subtiles) each run a 16-step v_wmma_f32_16x16x4_f32 chain over c=64.
// Epilogue: + u1[k] + v1[l], diag override, ReLU.
// ---------------------------------------------------------------------------
__global__ __launch_bounds__(256) void layer1_wmma_k(
    const float* __restrict__ A1, const float* __restrict__ Wf,
    const float* __restrict__ dp1, const float* __restrict__ u1,
    const float* __restrict__ v1, float* __restrict__ A2) {
  __shared__ float Ws[CH * WS_STRIDE];   // 17408 B
  __shared__ float Xs[CH * XS_STRIDE];   // 10240 B

  int b = blockIdx.y;
  int k = blockIdx.x >> 4;                 // row index 0..511
  int l0blk = (blockIdx.x & 15) * 32;      // 32-wide column chunk
  int tid = threadIdx.x;
  int lane = tid & 31;
  int w = tid >> 5;
  int ot = w & 3;                          // o-tile 0..3
  int nt = w >> 2;                         // n-subtile 0..1
  int obase = ot * 16;
  int lane15 = lane & 15;
  int hi = (lane >> 4) & 1;

  const float* Xg = A1 + (size_t)b * CH * KK + (size_t)k * KDIM + l0blk;

  // ---- stage W: 64x64 floats = 1024 x 16B chunks, 4 per thread ----
#pragma unroll
  for (int i = tid; i < CH * 16; i += 256) {
    int o = i >> 4, q = i & 15;
#if ASYNC_LDS
    __builtin_amdgcn_global_load_async_to_lds_b128(
        (gv4i_p)(Wf + o * CH + q * 4),
        (lv4i_p)(Ws + o * WS_STRIDE + q * 4), 0, 0);
#else
    *(v4f*)&Ws[o * WS_STRIDE + q * 4] = *(const v4f*)(Wf + o * CH + q * 4);
#endif
  }
  // ---- stage X: 64x32 floats = 512 x 16B chunks, 2 per thread ----
#pragma unroll
  for (int i = tid; i < CH * 8; i += 256) {
    int c = i >> 3, q = i & 7;
#if ASYNC_LDS
    __builtin_amdgcn_global_load_async_to_lds_b128(
        (gv4i_p)(Xg + (size_t)c * KK + q * 4),
        (lv4i_p)(Xs + c * XS_STRIDE + q * 4), 0, 0);
#else
    *(v4f*)&Xs[c * XS_STRIDE + q * 4] = *(const v4f*)(Xg + (size_t)c * KK + q * 4);
#endif
  }

  // prefetch epilogue operands while the DMA runs
  const float* ub  = u1  + (size_t)b * CH * KDIM;
  const float* vb  = v1  + (size_t)b * CH * KDIM;
  const float* dpb = dp1 + (size_t)b * CH * KDIM;
  __builtin_prefetch(ub + obase * KDIM + k, 0, 3);
  __builtin_prefetch(vb + obase * KDIM + l0blk, 0, 3);
  __builtin_prefetch(dpb + obase * KDIM + k, 0, 3);

#if ASYNC_LDS
  __builtin_amdgcn_s_wait_asynccnt(0);
#endif
  __syncthreads();

  int ncol = nt * 16 + lane15;             // column within the 32-wide chunk
  v8f acc = {};
#pragma unroll
  for (int s = 0; s < 16; ++s) {
    int ca = s * 4 + hi * 2;               // K-slice: lanes 0-15 get c0/c0+1, 16-31 get c0+2/c0+3
    v2f a, bb;
    a.x = Ws[(obase + lane15) * WS_STRIDE + ca];
    a.y = Ws[(obase + lane15) * WS_STRIDE + ca + 1];
    bb.x = Xs[ca * XS_STRIDE + ncol];
    bb.y = Xs[(ca + 1) * XS_STRIDE + ncol];
    // 8 args: (neg_a, A, neg_b, B, c_mod, C, reuse_a, reuse_b)
    acc = __builtin_amdgcn_wmma_f32_16x16x4_f32(false, a, false, bb,
                                                (short)0, acc, false, false);
  }

  int l = l0blk + ncol;
#pragma unroll
  for (int r = 0; r < 8; ++r) {
    int o = obase + r + hi * 8;            // C/D layout: VGPR r = rows r (lo half) / r+8 (hi half)
    float val = acc[r] + ub[o * KDIM + k] + vb[o * KDIM + l];
    if (l == k) val = dpb[o * KDIM + k];
    val = val > 0.f ? val : 0.f;
    A2[((size_t)b * CH + o) * KK + (size_t)k * KDIM + l] = val;
  }
}

// ---------------------------------------------------------------------------
// Layer 2 + readout, entirely from rs2/cs2/dg2 (no KxK tensor materialized)
// ---------------------------------------------------------------------------
__device__ inline float block_reduce_512(float x, float* sh, int k) {
  sh[k] = x; __syncthreads();
  for (int st = 256; st > 0; st >>= 1) { if (k < st) sh[k] += sh[k + st]; __syncthreads(); }
  float r = sh[0]; __syncthreads();
  return r;
}

__global__ __launch_bounds__(512) void final_k(
    const float* __restrict__ P1, const float* __restrict__ P2,
    const float* __restrict__ P3, const float* __restrict__ P4,
    const float* __restrict__ P5, const float* __restrict__ P6,
    const float* __restrict__ P7, const float* __restrict__ P8,
    const float* __restrict__ rs, const float* __restrict__ cs,
    const float* __restrict__ dg,
    const float* __restrict__ s1, const float* __restrict__ t1,
    const float* __restrict__ Q, float* __restrict__ out) {
  __shared__ float sh[512];
  int b = blockIdx.x;
  int k = threadIdx.x;
  float dp = 0.f, u = 0.f, v = 0.f, gr = 0.f, gc = 0.f;
  for (int c = 0; c < CH; ++c) {
    float sc = s1[c], tc = t1[c];
    size_t base = ((size_t)b * CH + c) * KDIM + k;
    float dn = sc * dg[base] + tc;
    float rn = sc * rs[base] + (float)KDIM * tc;
    float cn = sc * cs[base] + (float)KDIM * tc;
    dp += P1[c] * dn + 0.1f * (P2[c] * rn + P3[c] * cn);
    u  += 0.1f * P6[c] * dn + 0.01f * P7[c] * rn;
    v  += 0.1f * P5[c] * dn + 0.01f * P8[c] * cn;
    gr += P4[c] * (rn - dn);
    gc += P4[c] * (cn - dn);
  }
  float Usum = block_reduce_512(u, sh, k);
  float Vsum = block_reduce_512(v, sh, k);
  float td   = block_reduce_512(dp, sh, k);
  float rs3 = gr + 511.f * u + (Vsum - v) + dp;   // row-sum of A3
  float cs3 = gc + (Usum - u) + 511.f * v + dp;   // col-sum of A3
  float ts = block_reduce_512(rs3, sh, k);
  float y = Q[0] * dp + 0.1f * Q[1] * rs3 + 0.1f * Q[2] * cs3
          + 0.05f * Q[3] * td + 0.01f * Q[4] * ts;
  out[b * KDIM + k] = 1.0f / (1.0f + __expf(-y));
}

// ---------------------------------------------------------------------------
extern "C" void kernel_launch(void* const* d_in, const int* in_sizes, int n_in,
                              void* d_out, int out_size, void* d_ws, size_t ws_size,
                              hipStream_t stream) {
  (void)in_sizes; (void)n_in; (void)out_size; (void)ws_size;
  const float* A = (const float*)d_in[0];
  const float* P[8][3];
  for (int p = 0; p < 8; ++p)
    for (int l = 0; l < 3; ++l)
      P[p][l] = (const float*)d_in[1 + p * 3 + l];
  const float* bw0 = (const float*)d_in[25];
  const float* bw1 = (const float*)d_in[26];
  const float* bb0 = (const float*)d_in[27];
  const float* bb1 = (const float*)d_in[28];
  const float* Q   = (const float*)d_in[29];
  float* out = (float*)d_out;

  float* ws = (float*)d_ws;
  size_t off = 0;
  const size_t BCKK = (size_t)BSZ * CH * KK;
  const size_t BCK  = (size_t)BSZ * CH * KDIM;
  float* A1  = ws + off; off += BCKK;
  float* A2  = ws + off; off += BCKK;
  float* rs1 = ws + off; off += BCK;
  float* cs1 = ws + off; off += BCK;
  float* dg1 = ws + off; off += BCK;
  float* rs2 = ws + off; off += BCK;
  float* cs2 = ws + off; off += BCK;
  float* dg2 = ws + off; off += BCK;
  float* dp0 = ws + off; off += BCK;
  float* u0  = ws + off; off += BCK;
  float* v0  = ws + off; off += BCK;
  float* dp1 = ws + off; off += BCK;
  float* u1  = ws + off; off += BCK;
  float* v1  = ws + off; off += BCK;
  float* sum0 = ws + off; off += BSZ * CH;
  float* sq0  = ws + off; off += BSZ * CH;
  float* sum1 = ws + off; off += BSZ * CH;
  float* sq1  = ws + off; off += BSZ * CH;
  float* s0 = ws + off; off += CH;
  float* t0 = ws + off; off += CH;
  float* s1 = ws + off; off += CH;
  float* t1 = ws + off; off += CH;
  float* rsA = ws + off; off += BSZ * KDIM;
  float* csA = ws + off; off += BSZ * KDIM;
  float* dgA = ws + off; off += BSZ * KDIM;
  float* Wf  = ws + off; off += CH * CH;

  reduceA_k<<<BSZ, KDIM, 0, stream>>>(A, rsA, csA, dgA);
  prep0_k<<<(int)(BCK / 256), 256, 0, stream>>>(P[0][0], P[1][0], P[2][0],
                                                P[4][0], P[5][0], P[6][0], P[7][0],
                                                rsA, csA, dgA, dp0, u0, v0);
  layer0_k<<<dim3(KDIM * 2, BSZ), 256, 0, stream>>>(A, P[3][0], dp0, u0, v0, A1);
  reduce_bc_k<<<dim3(CH, BSZ), 256, 0, stream>>>(A1, rs1, cs1, dg1, sum0, sq0);
  bn_fin_k<<<1, CH, 0, stream>>>(sum0, sq0, bw0, bb0, s0, t0);
  prep1_k<<<(int)(BCK / 256), 256, 0, stream>>>(P[0][1], P[1][1], P[2][1], P[3][1],
                                                P[4][1], P[5][1], P[6][1], P[7][1],
                                                rs1, cs1, dg1, s0, t0, dp1, u1, v1);
  foldW_k<<<(CH * CH + 255) / 256, 256, 0, stream>>>(P[3][1], s0, Wf);
  layer1_wmma_k<<<dim3(KDIM * (KDIM / 32), BSZ), 256, 0, stream>>>(A1, Wf, dp1, u1, v1, A2);
  reduce_bc_k<<<dim3(CH, BSZ), 256, 0, stream>>>(A2, rs2, cs2, dg2, sum1, sq1);
  bn_fin_k<<<1, CH, 0, stream>>>(sum1, sq1, bw1, bb1, s1, t1);
  final_k<<<BSZ, KDIM, 0, stream>>>(P[0][2], P[1][2], P[2][2], P[3][2],
                                    P[4][2], P[5][2], P[6][2], P[7][2],
                                    rs2, cs2, dg2, s1, t1, Q, out);
}